// MambaLayer_1915555414297
// MI455X (gfx1250) — compile-verified
//
#include <hip/hip_runtime.h>
#include <hip/hip_bf16.h>
#include <math.h>
#include <stdint.h>

typedef float v2f __attribute__((ext_vector_type(2)));
typedef float v8f __attribute__((ext_vector_type(8)));

#define D_MODEL   512
#define D_STATE   64
#define HEADDIM   64
#define D_INNER   1024
#define NHEADS    16
#define CONV_CH   1152   // D_INNER + 2*D_STATE
#define D_IN_PROJ 2192   // 2*D_INNER + 2*D_STATE + NHEADS
#define BATCH     2
#define SEQLEN    2048
#define MROWS     4096   // BATCH*SEQLEN

#define KC        64     // K-chunk staged in LDS per double-buffer slot
#define WPAD      68     // padded row stride (floats): 272B = 17*16 -> B128-aligned,
                         // read bank = (4*r + kk) % 64 -> conflict-free

// ---------------------------------------------------------------------------
// fp32 WMMA GEMM with async-to-LDS staging of the shared W strip.
//   C[M,N] = op( A[M,K] * W[N,K]^T )
//   mode 0: C = acc
//   mode 1: C = relu(acc + bias[col]) + resid[row,col]   (MLP epilogue)
// Block = 256 threads (8 waves) -> 512 rows x 16 cols of C.
// All 8 waves share one 16-row W strip: staged via
// GLOBAL_LOAD_ASYNC_TO_LDS_B128 (ASYNCcnt), double-buffered.
// grid = (N/16, M/512).
// ---------------------------------------------------------------------------
__global__ __launch_bounds__(256) void gemm_f32_wmma(
    const float* __restrict__ A, const float* __restrict__ W,
    float* __restrict__ C, int M, int N, int K,
    const float* __restrict__ bias, const float* __restrict__ resid, int mode)
{
    __shared__ float Wlds[2][16 * WPAD];   // 2 x 4352 B

    const int tid  = threadIdx.x;
    const int lane = tid & 31;
    const int wave = tid >> 5;
    const int col0 = blockIdx.x * 16;
    const int row0 = (blockIdx.y * 8 + wave) * 64;     // 4 M-tiles of 16 rows
    const int half = lane >> 4;                        // 0: K+0..1, 1: K+2..3
    const int r    = lane & 15;

    // W staging decomposition: thread t async-copies 4 consecutive floats
    const int wr = tid >> 4;          // 0..15 : W row within the strip
    const int wc = (tid & 15) * 4;    // 0..60 : float col within chunk
    const float* __restrict__ wsrc = W + (size_t)(col0 + wr) * K + wc;

    const float* __restrict__ ar0 = A + (size_t)(row0 + r) * K;
    const float* __restrict__ ar1 = ar0 + (size_t)16 * K;
    const float* __restrict__ ar2 = ar0 + (size_t)32 * K;
    const float* __restrict__ ar3 = ar0 + (size_t)48 * K;

    // kick off chunk 0 fill (ASYNCcnt++)
    {
        uint32_t dst = (uint32_t)(uintptr_t)&Wlds[0][wr * WPAD + wc];
        asm volatile("global_load_async_to_lds_b128 %0, %1, off"
                     :: "v"(dst), "v"(wsrc) : "memory");
    }

    v8f acc0 = {}, acc1 = {}, acc2 = {}, acc3 = {};

    const int nch = K / KC;
    for (int c = 0; c < nch; ++c) {
        // wait for this chunk's async fill, then rendezvous.
        asm volatile("s_wait_asynccnt 0" ::: "memory");
        __syncthreads();

        if (c + 1 < nch) {
            // overlap: fill the other buffer while we compute on this one
            uint32_t dst = (uint32_t)(uintptr_t)&Wlds[(c + 1) & 1][wr * WPAD + wc];
            const float* src = wsrc + (size_t)(c + 1) * KC;
            asm volatile("global_load_async_to_lds_b128 %0, %1, off"
                         :: "v"(dst), "v"(src) : "memory");
            // prefetch next A chunk (global_prefetch_b8)
            const int kn = (c + 1) * KC;
            __builtin_prefetch(ar0 + kn, 0, 0);
            __builtin_prefetch(ar1 + kn, 0, 0);
            __builtin_prefetch(ar2 + kn, 0, 0);
            __builtin_prefetch(ar3 + kn, 0, 0);
        }

        const float* __restrict__ lw = &Wlds[c & 1][r * WPAD];
        const int k0 = c * KC;
        #pragma unroll 4
        for (int kk2 = 0; kk2 < KC; kk2 += 4) {
            const int kk = kk2 + 2 * half;
            v2f b;  b.x  = lw[kk];       b.y  = lw[kk + 1];       // ds_load_b64
            const int kg = k0 + kk;
            v2f a0; a0.x = ar0[kg];      a0.y = ar0[kg + 1];
            v2f a1; a1.x = ar1[kg];      a1.y = ar1[kg + 1];
            v2f a2; a2.x = ar2[kg];      a2.y = ar2[kg + 1];
            v2f a3; a3.x = ar3[kg];      a3.y = ar3[kg + 1];
            acc0 = __builtin_amdgcn_wmma_f32_16x16x4_f32(false, a0, false, b, (short)0, acc0, false, false);
            acc1 = __builtin_amdgcn_wmma_f32_16x16x4_f32(false, a1, false, b, (short)0, acc1, false, false);
            acc2 = __builtin_amdgcn_wmma_f32_16x16x4_f32(false, a2, false, b, (short)0, acc2, false, false);
            acc3 = __builtin_amdgcn_wmma_f32_16x16x4_f32(false, a3, false, b, (short)0, acc3, false, false);
        }
    }

    // C/D layout: VGPR i -> row (i + 8*half), col = r within the 16x16 tile
    const int col = col0 + r;
    if (mode == 0) {
        #pragma unroll
        for (int i = 0; i < 8; ++i) {
            C[(size_t)(row0 +  0 + i + 8*half) * N + col] = acc0[i];
            C[(size_t)(row0 + 16 + i + 8*half) * N + col] = acc1[i];
            C[(size_t)(row0 + 32 + i + 8*half) * N + col] = acc2[i];
            C[(size_t)(row0 + 48 + i + 8*half) * N + col] = acc3[i];
        }
    } else {
        const float bb = bias[col];
        #pragma unroll
        for (int i = 0; i < 8; ++i) {
            int rw0 = row0 +  0 + i + 8*half;
            int rw1 = row0 + 16 + i + 8*half;
            int rw2 = row0 + 32 + i + 8*half;
            int rw3 = row0 + 48 + i + 8*half;
            float h0 = acc0[i] + bb; h0 = h0 > 0.f ? h0 : 0.f;
            float h1 = acc1[i] + bb; h1 = h1 > 0.f ? h1 : 0.f;
            float h2 = acc2[i] + bb; h2 = h2 > 0.f ? h2 : 0.f;
            float h3 = acc3[i] + bb; h3 = h3 > 0.f ? h3 : 0.f;
            C[(size_t)rw0 * N + col] = h0 + resid[(size_t)rw0 * N + col];
            C[(size_t)rw1 * N + col] = h1 + resid[(size_t)rw1 * N + col];
            C[(size_t)rw2 * N + col] = h2 + resid[(size_t)rw2 * N + col];
            C[(size_t)rw3 * N + col] = h3 + resid[(size_t)rw3 * N + col];
        }
    }
}

// ---------------------------------------------------------------------------
// Depthwise causal conv (k=4) + bias + SiLU over xBC channels.
// ---------------------------------------------------------------------------
__global__ void conv_silu_kernel(const float* __restrict__ zxbcdt,
                                 const float* __restrict__ conv_w,
                                 const float* __restrict__ conv_b,
                                 float* __restrict__ xBC)
{
    int idx = blockIdx.x * blockDim.x + threadIdx.x;   // over MROWS*CONV_CH
    if (idx >= MROWS * CONV_CH) return;
    int c  = idx % CONV_CH;
    int bl = idx / CONV_CH;
    int b  = bl / SEQLEN;
    int l  = bl % SEQLEN;
    const float* w = conv_w + c * 4;
    float acc = conv_b[c];
    #pragma unroll
    for (int j = 0; j < 4; ++j) {
        int ls = l - 3 + j;
        if (ls >= 0)
            acc += zxbcdt[(size_t)(b * SEQLEN + ls) * D_IN_PROJ + D_INNER + c] * w[j];
    }
    float s = acc / (1.f + __expf(-acc));              // SiLU
    xBC[(size_t)bl * CONV_CH + c] = s;
}

// ---------------------------------------------------------------------------
// dt = softplus(raw + dt_bias)
// ---------------------------------------------------------------------------
__global__ void dt_kernel(const float* __restrict__ zxbcdt,
                          const float* __restrict__ dt_bias,
                          float* __restrict__ dt)
{
    int idx = blockIdx.x * blockDim.x + threadIdx.x;   // over MROWS*NHEADS
    if (idx >= MROWS * NHEADS) return;
    int hh = idx & (NHEADS - 1);
    int bl = idx >> 4;
    float v = zxbcdt[(size_t)bl * D_IN_PROJ + (D_IN_PROJ - NHEADS) + hh] + dt_bias[hh];
    dt[idx] = (v > 20.f) ? v : log1pf(__expf(v));
}

// ---------------------------------------------------------------------------
// Sequential SSM scan: one block per (batch, head). 256 threads, each owns
// 16 of the 64x64 state elems in registers; B/C staged in LDS per step.
// ---------------------------------------------------------------------------
__global__ __launch_bounds__(256) void ssm_scan_kernel(
    const float* __restrict__ xBC, const float* __restrict__ dt,
    const float* __restrict__ A_log, const float* __restrict__ D_skip,
    float* __restrict__ y)
{
    const int bh = blockIdx.x;              // 0..31
    const int b  = bh >> 4, hh = bh & 15;
    const int t  = threadIdx.x;
    const int p  = t >> 2;                  // 0..63  (HEADDIM index)
    const int n0 = (t & 3) << 4;            // 16-wide slice of D_STATE

    const float Aneg = -__expf(A_log[hh]);
    const float Dsk  = D_skip[hh];

    float hstate[16];
    #pragma unroll
    for (int i = 0; i < 16; ++i) hstate[i] = 0.f;

    __shared__ float Bs[D_STATE];
    __shared__ float Cs[D_STATE];

    const size_t rowBC = (size_t)b * SEQLEN * CONV_CH;
    const size_t rowdt = (size_t)b * SEQLEN * NHEADS;
    const size_t rowy  = (size_t)b * SEQLEN * D_INNER;

    for (int l = 0; l < SEQLEN; ++l) {
        const float* xb = xBC + rowBC + (size_t)l * CONV_CH;
        if (t < 64)       Bs[t]      = xb[D_INNER + t];
        else if (t < 128) Cs[t - 64] = xb[D_INNER + t];   // 1088..1151
        __syncthreads();

        const float dtv = dt[rowdt + (size_t)l * NHEADS + hh];
        const float dA  = __expf(dtv * Aneg);
        const float xp  = xb[hh * HEADDIM + p];
        const float dtx = dtv * xp;

        float yacc = 0.f;
        #pragma unroll
        for (int i = 0; i < 16; ++i) {
            float hn = hstate[i] * dA + dtx * Bs[n0 + i];
            hstate[i] = hn;
            yacc += hn * Cs[n0 + i];
        }
        yacc += __shfl_xor(yacc, 1);
        yacc += __shfl_xor(yacc, 2);
        if ((t & 3) == 0)
            y[rowy + (size_t)l * D_INNER + hh * HEADDIM + p] = yacc + Dsk * xp;
        __syncthreads();
    }
}

// ---------------------------------------------------------------------------
// y = rmsnorm(y * silu(z), norm_w, 1e-5)   in place, one block per row
// ---------------------------------------------------------------------------
__global__ __launch_bounds__(256) void gate_rmsnorm_kernel(
    const float* __restrict__ zxbcdt, const float* __restrict__ norm_w,
    float* __restrict__ y)
{
    const int row = blockIdx.x;
    const int t   = threadIdx.x;
    const float* zr = zxbcdt + (size_t)row * D_IN_PROJ;
    float* yr = y + (size_t)row * D_INNER;

    float g[4]; float ss = 0.f;
    #pragma unroll
    for (int j = 0; j < 4; ++j) {
        int idx = t + j * 256;
        float z = zr[idx];
        float s = z / (1.f + __expf(-z));
        float v = yr[idx] * s;
        g[j] = v; ss += v * v;
    }
    __shared__ float red[256];
    red[t] = ss; __syncthreads();
    for (int off = 128; off > 0; off >>= 1) {
        if (t < off) red[t] += red[t + off];
        __syncthreads();
    }
    const float inv = rsqrtf(red[0] / (float)D_INNER + 1e-5f);
    #pragma unroll
    for (int j = 0; j < 4; ++j) {
        int idx = t + j * 256;
        yr[idx] = norm_w[idx] * g[j] * inv;
    }
}

// ---------------------------------------------------------------------------
// out = rmsnorm(in, w, 1e-6), rows of 512, one block per row
// ---------------------------------------------------------------------------
__global__ __launch_bounds__(256) void rmsnorm512_kernel(
    const float* __restrict__ in, const float* __restrict__ w,
    float* __restrict__ out)
{
    const int row = blockIdx.x;
    const int t   = threadIdx.x;
    const float* ir = in + (size_t)row * D_MODEL;
    float a0 = ir[t], a1 = ir[t + 256];
    __shared__ float red[256];
    red[t] = a0 * a0 + a1 * a1; __syncthreads();
    for (int off = 128; off > 0; off >>= 1) {
        if (t < off) red[t] += red[t + off];
        __syncthreads();
    }
    const float inv = rsqrtf(red[0] / (float)D_MODEL + 1e-6f);
    out[(size_t)row * D_MODEL + t]       = w[t]       * a0 * inv;
    out[(size_t)row * D_MODEL + t + 256] = w[t + 256] * a1 * inv;
}

// ---------------------------------------------------------------------------
// Workspace layout (floats)
// ---------------------------------------------------------------------------
static const size_t OFF_ZX  = 0;                                        // 4096*2192
static const size_t OFF_XBC = OFF_ZX  + (size_t)MROWS * D_IN_PROJ;      // 4096*1152
static const size_t OFF_DT  = OFF_XBC + (size_t)MROWS * CONV_CH;        // 4096*16
static const size_t OFF_Y   = OFF_DT  + (size_t)MROWS * NHEADS;         // 4096*1024
static const size_t OFF_XR  = OFF_Y   + (size_t)MROWS * D_INNER;        // 4096*512
static const size_t OFF_H   = OFF_XR  + (size_t)MROWS * D_MODEL;        // 4096*512

extern "C" void kernel_launch(void* const* d_in, const int* in_sizes, int n_in,
                              void* d_out, int out_size, void* d_ws, size_t ws_size,
                              hipStream_t stream)
{
    const float* x          = (const float*)d_in[0];
    const float* in_proj_w  = (const float*)d_in[1];
    const float* conv_w     = (const float*)d_in[2];
    const float* conv_b     = (const float*)d_in[3];
    const float* dt_bias    = (const float*)d_in[4];
    const float* A_log      = (const float*)d_in[5];
    const float* D_skip     = (const float*)d_in[6];
    const float* norm_w     = (const float*)d_in[7];
    const float* out_proj_w = (const float*)d_in[8];
    const float* norm2_w    = (const float*)d_in[9];
    const float* mlp_w      = (const float*)d_in[10];
    const float* mlp_b      = (const float*)d_in[11];

    float* ws     = (float*)d_ws;
    float* zxbcdt = ws + OFF_ZX;
    float* xBC    = ws + OFF_XBC;
    float* dt     = ws + OFF_DT;
    float* ybuf   = ws + OFF_Y;
    float* xr     = ws + OFF_XR;
    float* hbuf   = ws + OFF_H;
    float* out    = (float*)d_out;

    // 1. in_proj GEMM: [4096,512] x [2192,512]^T -> [4096,2192]
    {
        dim3 grid(D_IN_PROJ / 16, MROWS / 512);
        gemm_f32_wmma<<<grid, 256, 0, stream>>>(x, in_proj_w, zxbcdt,
                                                MROWS, D_IN_PROJ, D_MODEL,
                                                nullptr, nullptr, 0);
    }
    // 2. depthwise conv + SiLU, and dt = softplus(raw + bias)
    {
        int total = MROWS * CONV_CH;
        conv_silu_kernel<<<(total + 255) / 256, 256, 0, stream>>>(zxbcdt, conv_w, conv_b, xBC);
        int tdt = MROWS * NHEADS;
        dt_kernel<<<(tdt + 255) / 256, 256, 0, stream>>>(zxbcdt, dt_bias, dt);
    }
    // 3. SSM scan (32 blocks = 2 batches x 16 heads)
    ssm_scan_kernel<<<BATCH * NHEADS, 256, 0, stream>>>(xBC, dt, A_log, D_skip, ybuf);

    // 4. gated RMSNorm in place on ybuf
    gate_rmsnorm_kernel<<<MROWS, 256, 0, stream>>>(zxbcdt, norm_w, ybuf);

    // 5. out_proj GEMM: [4096,1024] x [512,1024]^T -> xr [4096,512]
    {
        dim3 grid(D_MODEL / 16, MROWS / 512);
        gemm_f32_wmma<<<grid, 256, 0, stream>>>(ybuf, out_proj_w, xr,
                                                MROWS, D_MODEL, D_INNER,
                                                nullptr, nullptr, 0);
    }
    // 6. RMSNorm(xr) -> hbuf
    rmsnorm512_kernel<<<MROWS, 256, 0, stream>>>(xr, norm2_w, hbuf);

    // 7. MLP GEMM + bias + ReLU + residual -> d_out
    {
        dim3 grid(D_MODEL / 16, MROWS / 512);
        gemm_f32_wmma<<<grid, 256, 0, stream>>>(hbuf, mlp_w, out,
                                                MROWS, D_MODEL, D_MODEL,
                                                mlp_b, xr, 1);
    }
}